// HdOVCModel_35201551958185
// MI455X (gfx1250) — compile-verified
//
#include <hip/hip_runtime.h>
#include <hip/hip_bf16.h>
#include <math.h>

typedef float v2f __attribute__((ext_vector_type(2)));
typedef float v8f __attribute__((ext_vector_type(8)));

#define PDIM 20000
#define NUM  256
#define LDIM 30
#define SLAB (NUM * LDIM)   // 7680 floats per p-slab, contiguous

__device__ __forceinline__ float period_bound_f(float d) {
    const float PI_F = 3.14159265358979323846f;
    float m = fmodf(d + PI_F, 2.0f * PI_F);
    if (m < 0.0f) m += 2.0f * PI_F;   // jnp mod: result in [0, 2pi)
    return m - PI_F;
}

// ---------------------------------------------------------------------------
// Kernel A: single block. Zero input_bearing accumulator, sum(r_hpc),
// first-occurrence argmax(r_hpc).
// ws_f[0..255] = input_bearing accumulator (zeroed)
// ws_f[256]    = sum(r_hpc)
// ws_i[257]    = argmax index
// ---------------------------------------------------------------------------
__global__ void __launch_bounds__(256) prep_kernel(const float* __restrict__ r_hpc,
                                                   float* __restrict__ ws_f,
                                                   int* __restrict__ ws_i) {
    __shared__ float s_sum[256];
    __shared__ float s_max[256];
    __shared__ int   s_idx[256];
    const int tid = threadIdx.x;
    ws_f[tid] = 0.0f;                       // zero einsum accumulator each call

    float sum = 0.0f, mx = -INFINITY; int mi = 0;
    for (int p = tid; p < PDIM; p += 256) {
        float v = r_hpc[p];
        sum += v;
        if (v > mx) { mx = v; mi = p; }     // strict > keeps first occurrence
    }
    s_sum[tid] = sum; s_max[tid] = mx; s_idx[tid] = mi;
    __syncthreads();
    for (int s = 128; s > 0; s >>= 1) {
        if (tid < s) {
            s_sum[tid] += s_sum[tid + s];
            if (s_max[tid + s] > s_max[tid] ||
                (s_max[tid + s] == s_max[tid] && s_idx[tid + s] < s_idx[tid])) {
                s_max[tid] = s_max[tid + s]; s_idx[tid] = s_idx[tid + s];
            }
        }
        __syncthreads();
    }
    if (tid == 0) { ws_f[256] = s_sum[0]; ws_i[257] = s_idx[0]; }
}

// ---------------------------------------------------------------------------
// Kernel B: the bandwidth kernel. For each p-slab (256x30 contiguous floats):
//   - accumulate ib[n] += norm_r[p] * sum_l CB[p,n,l]*r_bearing[l] via
//     V_WMMA_F32_16X16X4_F32 (A = 16x4 CB tile, B = broadcast weights)
//   - write the slab to d_out (fused copy; saves a full 614MB read pass)
// A-matrix 16x4 f32 VGPR layout (ISA 7.12.2): lane<16 holds K=l0,l0+1 in
// v0,v1; lane>=16 holds K=l0+2,l0+3  ->  one aligned float2 load per lane.
// ---------------------------------------------------------------------------
__global__ void __launch_bounds__(256) einsum_copy_kernel(
        const float* __restrict__ cb, const float* __restrict__ r_hpc,
        const float* __restrict__ r_bearing, float* __restrict__ ws_f,
        float* __restrict__ out_cb) {
    __shared__ float w_lds[32];            // w[l] = r_bearing[l]*norm_r[p]; [30],[31]=0
    const int tid  = threadIdx.x;
    const int lane = tid & 31;
    const int wave = tid >> 5;             // 0..7
    const int half = lane >> 4;            // 0/1  (K split)
    const int mrow = lane & 15;            // M within tile
    const int nb0  = wave * 32;            // this wave's first 16-row tile
    const int nb1  = nb0 + 16;             // second tile
    const float invSum = 1.0f / ws_f[256];

    v8f acc0 = {}; v8f acc1 = {};

    for (int p = blockIdx.x; p < PDIM; p += gridDim.x) {
        const float nr = r_hpc[p] * invSum;
        __syncthreads();
        if (tid < 32) w_lds[tid] = (tid < LDIM) ? (r_bearing[tid] * nr) : 0.0f;
        __syncthreads();

        const float* slab  = cb     + (size_t)p * SLAB;
        float*       oslab = out_cb + (size_t)p * SLAB;

        // software-pipeline the next slab (global_prefetch_b8); 256 lanes at
        // 120B spacing cover all 240 cachelines of the 30KB slab
        const int pn = p + gridDim.x;
        if (pn < PDIM) __builtin_prefetch(cb + (size_t)pn * SLAB + tid * LDIM, 0, 1);

        #pragma unroll
        for (int l0 = 0; l0 < 32; l0 += 4) {          // 8 K-chunks of 4
            const int l  = l0 + 2 * half;             // this half's first K
            const int lc = (l < LDIM) ? l : (LDIM - 2);  // clamp tail (w=0 there)

            v2f B; B.x = w_lds[l]; B.y = w_lds[l + 1];

            const v2f a0 = *(const v2f*)(slab + (size_t)(nb0 + mrow) * LDIM + lc);
            const v2f a1 = *(const v2f*)(slab + (size_t)(nb1 + mrow) * LDIM + lc);

#if defined(__has_builtin) && __has_builtin(__builtin_amdgcn_wmma_f32_16x16x4_f32)
            acc0 = __builtin_amdgcn_wmma_f32_16x16x4_f32(false, a0, false, B,
                                                         (short)0, acc0, false, false);
            acc1 = __builtin_amdgcn_wmma_f32_16x16x4_f32(false, a1, false, B,
                                                         (short)0, acc1, false, false);
#else
            acc0[0] += a0.x * B.x + a0.y * B.y;       // fallback (per-lane dot)
            acc1[0] += a1.x * B.x + a1.y * B.y;
#endif
            if (l < LDIM) {                           // fused copy to output
                *(v2f*)(oslab + (size_t)(nb0 + mrow) * LDIM + l) = a0;
                *(v2f*)(oslab + (size_t)(nb1 + mrow) * LDIM + l) = a1;
            }
        }
    }

    // Flush C tiles. C layout: VGPR i, lanes 0-15 -> M=i ; lanes 16-31 -> M=8+i.
    // All 16 columns are identical (B was column-broadcast), so lane 0 and
    // lane 16 carry the full column.
#if defined(__has_builtin) && __has_builtin(__builtin_amdgcn_wmma_f32_16x16x4_f32)
    if (lane == 0) {
        #pragma unroll
        for (int i = 0; i < 8; ++i) {
            atomicAdd(&ws_f[nb0 + i], acc0[i]);
            atomicAdd(&ws_f[nb1 + i], acc1[i]);
        }
    } else if (lane == 16) {
        #pragma unroll
        for (int i = 0; i < 8; ++i) {
            atomicAdd(&ws_f[nb0 + 8 + i], acc0[i]);
            atomicAdd(&ws_f[nb1 + 8 + i], acc1[i]);
        }
    }
#else
    atomicAdd(&ws_f[nb0 + mrow], acc0[0]);
    atomicAdd(&ws_f[nb1 + mrow], acc1[0]);
#endif
}

// ---------------------------------------------------------------------------
// Kernel C: single block of 256 (one thread per neuron n). All the small math.
// ---------------------------------------------------------------------------
__device__ __forceinline__ float blk_reduce_sum(float val, float* buf) {
    const int n = threadIdx.x;
    buf[n] = val; __syncthreads();
    for (int s = 128; s > 0; s >>= 1) {
        if (n < s) buf[n] += buf[n + s];
        __syncthreads();
    }
    float res = buf[0];
    __syncthreads();
    return res;
}

__global__ void __launch_bounds__(256) finalize_kernel(
        const float* __restrict__ cb, const float* __restrict__ r,
        const float* __restrict__ u, const float* __restrict__ v,
        const float* __restrict__ r_bearing, const float* __restrict__ pos_estimate,
        const float* __restrict__ ang, const float* __restrict__ sen2,
        const float* __restrict__ shared_t, const float* __restrict__ noise_hd,
        const float* __restrict__ noise_bearing, const float* __restrict__ dtp,
        const float* __restrict__ ws_f, const int* __restrict__ ws_i,
        float* __restrict__ out) {
    const int n = threadIdx.x;
    __shared__ float buf[256];
    __shared__ float r_s[256];
    __shared__ float conn_s[256];
    __shared__ float rh_s[256];
    __shared__ float bufv[256];
    __shared__ int   bufi[256];
    __shared__ float rbp_s[32];

    const float PI_F = 3.14159265358979323846f;
    const float GW = PI_F * 0.25f, GH = 3.0f;
    const float TAU = 1.0f, TAU_V = 10.0f, TAU_CONN = 10.0f, TAU_E = 1000.0f;
    const float LR = 10000.0f;
    const float mconst = 0.84f * TAU / TAU_V;          // 0.084
    const float kconst = 0.8f / NUM * 20.0f;           // 0.0625
    const float J0     = 1.0f / NUM * 20.0f;           // 0.078125

    const float dx = 2.0f * PI_F / NUM;
    const float x  = -PI_F + dx * n;
    const float rn = r[n];
    r_s[n] = rn;

    // ring convolution kernel conn[i] (replaces fft/ifft pair)
    {
        const float d0 = period_bound_f(fabsf(-PI_F - x));
        const float t  = d0 / GW;
        conn_s[n] = J0 * expf(-0.5f * t * t) / (2.0f * PI_F * GW * GW);
    }
    if (n < LDIM) rbp_s[n] = r_bearing[n] + fabsf(noise_bearing[n]);
    __syncthreads();

    // center of mass on the ring
    const float sumSin = blk_reduce_sum(sinf(x) * rn, buf);
    const float sumCos = blk_reduce_sum(cosf(x) * rn, buf);
    const float center = atan2f(sumSin, sumCos);

    const float dt   = dtp[0];
    const float pos0 = pos_estimate[0];
    const float pos_e = pos0 + (period_bound_f(center - pos0) / TAU_E + ang[0]) * dt;
    const float pos_est = period_bound_f(pos_e);
    const float md = period_bound_f(x - pos_est) / GW;
    const float motion = GH * expf(-0.25f * md * md);

    // Irec: 256-point circular convolution conn (*) r
    float Irec = 0.0f;
    for (int j = 0; j < NUM; ++j)
        Irec += conn_s[(n - j) & (NUM - 1)] * r_s[j];

    const float input_total = motion + Irec + ws_f[n] * sen2[0];

    // exact exponential-Euler CANN update
    const float t  = shared_t[1] - shared_t[0];
    const float un = u[n], vn = v[n];
    const float Cu = input_total - vn;
    const float Cv = mconst * un;
    const float u_sol = Cu + (un - Cu) * expf(-t / TAU);
    const float v_sol = Cv + (vn - Cv) * expf(-t / TAU_V);
    const float u_new = fmaxf(u_sol, 0.0f);
    const float r1 = u_new * u_new;
    const float sum_r1 = blk_reduce_sum(r1, buf);
    const float r_new = r1 / (1.0f + kconst * sum_r1);

    out[n]       = r_new;
    out[256 + n] = u_new;
    out[512 + n] = v_sol;

    // top-4 threshold of r_hd (duplicates counted, like top_k)
    const float r_hd = r_new * (1.0f + fabsf(noise_hd[n]));
    rh_s[n] = r_hd;
    __syncthreads();
    float thr = 0.0f;
    for (int it = 0; it < 4; ++it) {
        bufv[n] = rh_s[n]; bufi[n] = n; __syncthreads();
        for (int s = 128; s > 0; s >>= 1) {
            if (n < s) {
                if (bufv[n + s] > bufv[n] ||
                    (bufv[n + s] == bufv[n] && bufi[n + s] < bufi[n])) {
                    bufv[n] = bufv[n + s]; bufi[n] = bufi[n + s];
                }
            }
            __syncthreads();
        }
        thr = bufv[0];
        if (n == 0) rh_s[bufi[0]] = -INFINITY;
        __syncthreads();
    }
    const float r_learn = (r_hd >= thr) ? r_hd : 0.0f;

    // Hebbian update + row-normalize of conn_bearing[index]; overwrite output row
    const int idx = ws_i[257];
    const float lrf = LR * dt / TAU_CONN;
    const float* rowIn  = cb  + (size_t)idx * SLAB + n * LDIM;
    float*       rowOut = out + 768 + (size_t)idx * SLAB + n * LDIM;
    float rowv[LDIM];
    float s = 0.0f;
    #pragma unroll
    for (int l = 0; l < LDIM; ++l) {
        const float c = rowIn[l] * (1.0f + lrf * r_learn * rbp_s[l]);
        rowv[l] = c; s += c;
    }
    const float inv = 1.0f / s;
    #pragma unroll
    for (int l = 0; l < LDIM; ++l) rowOut[l] = rowv[l] * inv;
}

// ---------------------------------------------------------------------------
extern "C" void kernel_launch(void* const* d_in, const int* in_sizes, int n_in,
                              void* d_out, int out_size, void* d_ws, size_t ws_size,
                              hipStream_t stream) {
    const float* cb            = (const float*)d_in[0];
    const float* r             = (const float*)d_in[1];
    const float* u             = (const float*)d_in[2];
    const float* v             = (const float*)d_in[3];
    const float* r_hpc         = (const float*)d_in[4];
    const float* r_bearing     = (const float*)d_in[5];
    const float* pos_estimate  = (const float*)d_in[6];
    const float* ang           = (const float*)d_in[7];
    // d_in[8] = HD (unused by the reference math)
    const float* sen2          = (const float*)d_in[9];
    const float* shared_t      = (const float*)d_in[10];
    const float* noise_hd      = (const float*)d_in[11];
    const float* noise_bearing = (const float*)d_in[12];
    const float* dtp           = (const float*)d_in[13];

    float* out = (float*)d_out;
    float* wsf = (float*)d_ws;
    int*   wsi = (int*)d_ws;

    prep_kernel<<<1, 256, 0, stream>>>(r_hpc, wsf, wsi);
    einsum_copy_kernel<<<2000, 256, 0, stream>>>(cb, r_hpc, r_bearing, wsf, out + 768);
    finalize_kernel<<<1, 256, 0, stream>>>(cb, r, u, v, r_bearing, pos_estimate,
                                           ang, sen2, shared_t, noise_hd,
                                           noise_bearing, dtp, wsf, wsi, out);
}